// CustomNLLLoss_20624432955694
// MI455X (gfx1250) — compile-verified
//
#include <hip/hip_runtime.h>
#include <hip/hip_bf16.h>

typedef float v2f __attribute__((ext_vector_type(2)));
typedef float v8f __attribute__((ext_vector_type(8)));

#define NTHREADS 256
#define NBLOCKS  2048

// Per-sample accumulate: 2-class log-softmax picked at target t (branchless).
__device__ __forceinline__ void acc_sample(float x0, float x1, int t,
                                           float& s0, float& s1, float& c1) {
    float m      = fmaxf(x0, x1);
    float lse    = m + __logf(1.0f + __expf(-fabsf(x0 - x1)));
    float picked = (t ? x1 : x0) - lse;
    float ft     = (float)t;                 // t in {0,1}
    s1 = fmaf(ft, picked, s1);
    s0 = fmaf(1.0f - ft, picked, s0);
    c1 += ft;
}

// Block-wide reduction of (s0,s1,c1) over 256 threads (8 wave32 waves).
// Intra-wave: shfl_xor tree. Cross-wave: one v_wmma_f32_16x16x4_f32 with an
// all-ones B matrix: A row r holds 4 wave-partials of accumulator r/2
// (rows 0..5 used), so D row-sums give the cross-wave totals.
// On return, thread 0 of the block holds the block totals.
__device__ __forceinline__ void block_reduce3(float& s0, float& s1, float& c1) {
    #pragma unroll
    for (int off = 16; off > 0; off >>= 1) {
        s0 += __shfl_xor(s0, off, 32);
        s1 += __shfl_xor(s1, off, 32);
        c1 += __shfl_xor(c1, off, 32);
    }
    __shared__ float part[8][3];
    const int lane = threadIdx.x & 31;
    const int wave = threadIdx.x >> 5;
    if (lane == 0) {
        part[wave][0] = s0; part[wave][1] = s1; part[wave][2] = c1;
    }
    __syncthreads();
    if (wave == 0) {
        // A-matrix (16x4 f32) layout: lane L<16 -> row L, K=0 (a.x), K=1 (a.y);
        // lane L>=16 -> row L-16, K=2 (a.x), K=3 (a.y).
        const int r    = lane & 15;
        const int half = lane >> 4;
        v2f a = {0.0f, 0.0f};
        if (r < 6) {
            const int ai = r >> 1;                       // accumulator index 0..2
            const int w  = ((r & 1) << 2) + (half << 1); // wave base 0/2/4/6
            a.x = part[w][ai];
            a.y = part[w + 1][ai];
        }
        v2f b = {1.0f, 1.0f};   // all-ones B: layout-invariant
        v8f c = {};
        v8f d = __builtin_amdgcn_wmma_f32_16x16x4_f32(
            /*neg_a=*/false, a, /*neg_b=*/false, b,
            /*c_mod=*/(short)0, c, /*reuse_a=*/false, /*reuse_b=*/false);
        // D layout: d[v] lanes 0..15 = rowsum(v). Rows 2a,2a+1 belong to acc a.
        s0 = d[0] + d[1];
        s1 = d[2] + d[3];
        c1 = d[4] + d[5];
    }
}

__global__ void __launch_bounds__(NTHREADS)
nll_partial_kernel(const float4* __restrict__ x4, const int4* __restrict__ t4,
                   float* __restrict__ partials, int n) {
    const int ng     = n >> 2;                          // groups of 4 samples
    const int tid    = blockIdx.x * blockDim.x + threadIdx.x;
    const int stride = gridDim.x * blockDim.x;

    float s0 = 0.0f, s1 = 0.0f, c1 = 0.0f;

    for (int g = tid; g < ng; g += stride) {
        // Prefetch next grid-stride tile (speculative; gfx1250 global_prefetch_b8).
        __builtin_prefetch(&x4[2 * (size_t)(g + stride)], 0, 0);
        __builtin_prefetch(&t4[(size_t)(g + stride)], 0, 0);

        float4 xa = x4[2 * (size_t)g];                  // samples 4g, 4g+1
        float4 xb = x4[2 * (size_t)g + 1];              // samples 4g+2, 4g+3
        int4   tt = t4[g];
        acc_sample(xa.x, xa.y, tt.x, s0, s1, c1);
        acc_sample(xa.z, xa.w, tt.y, s0, s1, c1);
        acc_sample(xb.x, xb.y, tt.z, s0, s1, c1);
        acc_sample(xb.z, xb.w, tt.w, s0, s1, c1);
    }

    // Scalar tail (n not multiple of 4): first few threads handle it.
    const int base = ng << 2;
    if (tid < (n - base)) {
        const float* xf = (const float*)x4;
        const int*   ti = (const int*)t4;
        const int    i  = base + tid;
        acc_sample(xf[2 * (size_t)i], xf[2 * (size_t)i + 1], ti[i], s0, s1, c1);
    }

    block_reduce3(s0, s1, c1);
    if (threadIdx.x == 0) {
        partials[blockIdx.x * 3 + 0] = s0;
        partials[blockIdx.x * 3 + 1] = s1;
        partials[blockIdx.x * 3 + 2] = c1;
    }
}

__global__ void __launch_bounds__(NTHREADS)
nll_final_kernel(const float* __restrict__ partials, int nb,
                 float* __restrict__ out, int n) {
    float s0 = 0.0f, s1 = 0.0f, c1 = 0.0f;
    for (int i = threadIdx.x; i < nb; i += blockDim.x) {
        s0 += partials[i * 3 + 0];
        s1 += partials[i * 3 + 1];
        c1 += partials[i * 3 + 2];
    }
    block_reduce3(s0, s1, c1);
    if (threadIdx.x == 0) {
        float n1 = c1;
        float n0 = (float)n - n1;
        float p  = (n0 > 0.0f) ? (-s0 / n0) : 0.0f;
        float r  = (n1 > 0.0f) ? (-s1 / n1) : 0.0f;
        out[0] = p + r;
    }
}

extern "C" void kernel_launch(void* const* d_in, const int* in_sizes, int n_in,
                              void* d_out, int out_size, void* d_ws, size_t ws_size,
                              hipStream_t stream) {
    (void)n_in; (void)out_size; (void)ws_size;
    const float* x  = (const float*)d_in[0];   // [N, 2] f32
    const int*   tg = (const int*)d_in[1];     // [N] targets (int)
    const int    n  = in_sizes[1];             // N samples

    float* partials = (float*)d_ws;            // [NBLOCKS][3] f32 = 24 KB

    nll_partial_kernel<<<NBLOCKS, NTHREADS, 0, stream>>>(
        (const float4*)x, (const int4*)tg, partials, n);
    nll_final_kernel<<<1, NTHREADS, 0, stream>>>(partials, NBLOCKS, (float*)d_out, n);
}